// BatchBlur_SV_4105988735742
// MI455X (gfx1250) — compile-verified
//
#include <hip/hip_runtime.h>
#include <stdint.h>

// Spatially-varying 19x19 blur, B=2, C=3, H=W=192, fp32.
// Memory-bound: ~106.5 MB per-pixel kernel stream dominates -> ~4.6us floor @23.3TB/s.
// CDNA5 paths: global_load_async_to_lds_b32 gather (ASYNCcnt) for the reflect-padded
// image tile, s_wait_asynccnt, global_prefetch_b8 on the kernel stream.
// WMMA intentionally not used: einsum npk,npk->np is a batched diagonal, not a GEMM;
// any 16x16 WMMA mapping wastes 16x compute on the slow f32 matrix path.

#define H_ 192
#define W_ 192
#define L_ 19
#define PAD_ 9
#define TILE_ 16
#define HALO_ 34              // TILE_ + 2*PAD_
#define RS_ 35                // padded LDS row stride (floats): conflict-free 2-rows/wave
#define PLANE_ (HALO_ * RS_)  // 1190 floats per channel plane

__device__ __forceinline__ int reflect192(int i) {
    i = (i < 0) ? -i : i;
    return (i >= H_) ? (2 * H_ - 2 - i) : i;
}

__global__ __launch_bounds__(256)
void sv_blur_kernel(const float* __restrict__ x,
                    const float* __restrict__ ker,
                    float* __restrict__ out)
{
    __shared__ float smem[3 * PLANE_];   // 3 channel planes, 34 rows x 35-float stride

    const int tid = threadIdx.x;
    const int tx  = tid & (TILE_ - 1);
    const int ty  = tid >> 4;
    const int bx0 = blockIdx.x * TILE_;
    const int by0 = blockIdx.y * TILE_;
    const int b   = (int)blockIdx.z;

    const unsigned lds_base = (unsigned)(uintptr_t)(void*)smem;

    // ---- Async gather of the reflect-padded 34x34x3 image tile into LDS ----
    // Thread decomposition 32x8 (shift/mask only, no integer division).
    // Per-lane LDS dest + per-lane global address = GLOBAL_LOAD_ASYNC_TO_LDS_B32.
    {
        const int lx = tid & 31;   // 0..31 column lane
        const int ly = tid >> 5;   // 0..7  row group
        #pragma unroll
        for (int c = 0; c < 3; ++c) {
            const float* plane = x + (size_t)(b * 3 + c) * (H_ * W_);
            for (int py = ly; py < HALO_; py += 8) {
                const int gy = reflect192(by0 + py - PAD_);
                const float* rowp = plane + (size_t)gy * W_;
                const unsigned ldrow =
                    lds_base + (unsigned)((c * PLANE_ + py * RS_) * 4);
                for (int px = lx; px < HALO_; px += 32) {
                    const int gx = reflect192(bx0 + px - PAD_);
                    const unsigned dst = ldrow + (unsigned)(px * 4);
                    asm volatile("global_load_async_to_lds_b32 %0, %1, off"
                                 :: "v"(dst), "v"((uint64_t)(uintptr_t)(rowp + gx))
                                 : "memory");
                }
            }
        }
    }
    asm volatile("s_wait_asynccnt 0" ::: "memory");
    __syncthreads();

    // ---- Per-pixel 361-tap MAC, one kernel-stream load reused over 3 channels ----
    const int h = by0 + ty;
    const int w = bx0 + tx;
    const float* kp = ker + ((size_t)b * (H_ * W_) + (size_t)(h * W_ + w)) * (L_ * L_);

    float a0 = 0.0f, a1 = 0.0f, a2 = 0.0f;

    for (int ki = 0; ki < L_; ++ki) {
        const float* krow = kp + ki * L_;
        // keep this thread's sequential kernel stream ahead of consumption
        __builtin_prefetch(krow + 2 * L_, 0, 1);   // -> global_prefetch_b8
        const int rowoff = (ty + ki) * RS_ + tx;
        #pragma unroll
        for (int kj = 0; kj < L_; ++kj) {
            const float kv  = krow[kj];
            const int   idx = rowoff + kj;
            a0 = fmaf(smem[idx],              kv, a0);
            a1 = fmaf(smem[PLANE_ + idx],     kv, a1);
            a2 = fmaf(smem[2 * PLANE_ + idx], kv, a2);
        }
    }

    float* o = out + (((size_t)(b * 3) * H_ + h) * W_ + w);
    o[0]                   = a0;
    o[(size_t)H_ * W_]     = a1;
    o[2 * (size_t)H_ * W_] = a2;
}

extern "C" void kernel_launch(void* const* d_in, const int* in_sizes, int n_in,
                              void* d_out, int out_size, void* d_ws, size_t ws_size,
                              hipStream_t stream)
{
    const float* x   = (const float*)d_in[0];  // (2,3,192,192) fp32
    const float* ker = (const float*)d_in[1];  // (2,36864,19,19) fp32
    float* out = (float*)d_out;                // (2,3,192,192) fp32

    dim3 grid(W_ / TILE_, H_ / TILE_, 2);
    sv_blur_kernel<<<grid, 256, 0, stream>>>(x, ker, out);
}